// V_theta_34626026340882
// MI455X (gfx1250) — compile-verified
//
#include <hip/hip_runtime.h>
#include <math.h>

// ===========================================================================
// MI455X (gfx1250) implementation of the equivariant GNN reference.
//
// Compute-bound, L2-resident problem (≈10 GFLOP, ≈50 MB of HBM traffic →
// 2 µs at 23.3 TB/s, so FLOPs/latency dominate). fp32 WMMA
// (v_wmma_f32_16x16x4_f32, K=16 via 4 chained ops) generates per-edge/bond
// tensor-product weights from the 16-wide MLP hidden state; CG contractions
// are fully unrolled VALU; scatter via global_atomic_add_f32. Wave32 blocks,
// padded LDS staging (bank-conflict free), B-fragments held in registers
// across the M-tile sweep, next-path weights prefetched (global_prefetch_b8).
// CG/path tables are rebuilt on-device every call (capture-safe).
// ===========================================================================

typedef float v2f __attribute__((ext_vector_type(2)));
typedef float v8f __attribute__((ext_vector_type(8)));

#define EPB 128   // edges per block (4 x wave32)
#define BPB 64    // bonds per block (2 x wave32)
#define HSTR 17   // padded LDS stride for 16-wide hidden state

// ---------------- problem constants ----------------
#define N_NODES 10000
#define N_EDGES 100000
#define N_BOND  1000
#define SCALING 0.2f

// ---------------- irreps descriptors: triples (mul, l, parity) -------------
__device__ const int IRD_IN[]   = {2,0,1};
__device__ const int IRD_SH[]   = {1,0,1, 1,1,-1, 1,2,1};
__device__ const int IRD_MID1[] = {8,0,1, 8,1,-1, 8,2,1};
__device__ const int IRD_MID2[] = {8,0,1, 8,0,-1, 8,1,1, 8,1,-1, 8,2,1, 8,2,-1};
__device__ const int IRD_HH[]   = {4,0,1, 2,1,-1, 2,1,-1, 1,0,1, 1,1,1, 1,2,1};
__device__ const int IRD_CC[]   = {
  9,0,1, 6,1,-1, 3,2,1, 6,1,-1,
  1,0,1, 1,1,1, 1,2,1,  1,0,1, 1,1,1, 1,2,1,
  1,0,1, 1,1,1, 1,2,1,  1,0,1, 1,1,1, 1,2,1,
  1,1,-1, 1,2,-1, 1,3,-1,  1,1,-1, 1,2,-1, 1,3,-1,
  3,2,1,
  1,1,-1, 1,2,-1, 1,3,-1,  1,1,-1, 1,2,-1, 1,3,-1,
  1,0,1, 1,1,1, 1,2,1, 1,3,1, 1,4,1};           // 34 slices, dim 196
__device__ const int IRD_CH[]   = {
  6,0,1, 3,1,-1, 4,1,-1,
  1,0,1, 1,1,1, 1,2,1,  1,0,1, 1,1,1, 1,2,1,
  2,2,1, 1,1,-1, 1,2,-1, 1,3,-1};               // 13 slices, dim 70
__device__ const int IRD_SCR[]  = {32,0,1, 1,2,1};
__device__ const int IRD_SCR2[] = {1,0,1, 1,2,1};
__device__ const int IRD_G1[]   = {32,0,1};
__device__ const int IRD_G2[]   = {3,0,1};

// ---------------- workspace layout (float words) ----------------
#define WS_CG   ((size_t)0)        // 27*125 CG tensors, fixed [5][5][5] stride
#define WS_P1   ((size_t)3392)     // TP path table 1 (ints)
#define WS_P2   ((size_t)3456)
#define WS_PB   ((size_t)3648)
#define WS_LT   ((size_t)4416)     // 10 linear pair tables x 512 ints
#define WS_X1   ((size_t)16384)    // 10000 x 2
#define WS_XN1  ((size_t)36384)    // 10000 x 72 (atomic accum)
#define WS_X2   ((size_t)756384)   // 10000 x 72
#define WS_XN2  ((size_t)1476384)  // 10000 x 144 (atomic accum)
#define WS_X3   ((size_t)2916384)  // 10000 x 144
#define WS_HHF  ((size_t)4356384)  // 1000 x 144
#define WS_CCF  ((size_t)4500384)
#define WS_CHF  ((size_t)4644384)
#define WS_SCR  ((size_t)4788384)  // 10000 x 37
#define WS_G1B  ((size_t)5158384)  // 1000 x 32

// ---------------- output layout (floats) ----------------
#define OUT_nH     ((size_t)0)        // 10000 x 25
#define OUT_nC     ((size_t)250000)   // 10000 x 196
#define OUT_eHH    ((size_t)2210000)  // 1000 x 25
#define OUT_eCH    ((size_t)2235000)  // 1000 x 70
#define OUT_eCC    ((size_t)2305000)  // 1000 x 196
#define OUT_screen ((size_t)2501000)  // 10000 x 6
#define OUT_gCC    ((size_t)2561000)  // 1000 x 3
#define OUT_gHH    ((size_t)2564000)
#define OUT_gCH    ((size_t)2567000)

// ===========================================================================
// Setup: Clebsch-Gordan (real basis) computed on device — exact port of the
// reference _cgc/_c2r.  T is purely real for even l1+l2+l3, imaginary for odd.
// ===========================================================================
__device__ double dfac(int n) { double r = 1.0; for (int i = 2; i <= n; i++) r *= i; return r; }

__device__ double cgc(int j1,int m1,int j2,int m2,int j3,int m3) {
  if (m1 + m2 != m3) return 0.0;
  double pre = sqrt((2.0*j3+1.0)*dfac(j3+j1-j2)*dfac(j3-j1+j2)*dfac(j1+j2-j3)/dfac(j1+j2+j3+1));
  pre *= sqrt(dfac(j3+m3)*dfac(j3-m3)*dfac(j1-m1)*dfac(j1+m1)*dfac(j2-m2)*dfac(j2+m2));
  int kmin = 0;
  if (j2-j3-m1 > kmin) kmin = j2-j3-m1;
  if (j1-j3+m2 > kmin) kmin = j1-j3+m2;
  int kmax = j1+j2-j3;
  if (j1-m1 < kmax) kmax = j1-m1;
  if (j2+m2 < kmax) kmax = j2+m2;
  double s = 0.0;
  for (int k = kmin; k <= kmax; k++) {
    double d = dfac(k)*dfac(j1+j2-j3-k)*dfac(j1-m1-k)*dfac(j2+m2-k)*dfac(j3-j2+m1+k)*dfac(j3-j1-m2+k);
    s += ((k & 1) ? -1.0 : 1.0) / d;
  }
  return pre * s;
}

// complex entry of the c2r matrix q[l]: row index mu (-l..l), column m (-l..l)
__device__ void qc2r(int mu, int m, double& zx, double& zy) {
  const double r2 = 0.70710678118654752440;
  zx = 0.0; zy = 0.0;
  if (mu == 0) { if (m == 0) zx = 1.0; }
  else if (mu > 0) {
    if (m == mu)       zx = (mu & 1) ? -r2 : r2;
    else if (m == -mu) zx = r2;
  } else {
    int n = -mu;
    if (m == mu)       zy = r2;             //  i*r2
    else if (m == n)   zy = (n & 1) ? r2 : -r2;  // -i*(-1)^n*r2
  }
}

__global__ void k_cg(float* __restrict__ cg) {
  int t = blockIdx.x * blockDim.x + threadIdx.x;
  if (t >= 27 * 125) return;
  int combo = t / 125, rem = t % 125;
  int l1 = combo / 9, l2 = (combo / 3) % 3, l3 = combo % 3;
  int a = rem / 25, b = (rem / 5) % 5, c = rem % 5;
  float outv = 0.f;
  int dmin = l1 - l2; if (dmin < 0) dmin = -dmin;
  if (l3 >= dmin && l3 <= l1 + l2 && a < 2*l1+1 && b < 2*l2+1 && c < 2*l3+1) {
    double re = 0.0, im = 0.0;
    for (int m = -l1; m <= l1; m++)
      for (int n = -l2; n <= l2; n++) {
        int o = m + n;
        if (o < -l3 || o > l3) continue;
        double cg0 = cgc(l1, m, l2, n, l3, o);
        if (cg0 == 0.0) continue;
        double q1x,q1y,q2x,q2y,q3x,q3y;
        qc2r(a - l1, m, q1x, q1y);
        qc2r(b - l2, n, q2x, q2y);
        qc2r(c - l3, o, q3x, q3y);   // conj applied below
        double zx = q1x*q2x - q1y*q2y;
        double zy = q1x*q2y + q1y*q2x;
        double rx = zx*q3x + zy*q3y;     // * conj(q3)
        double ry = zy*q3x - zx*q3y;
        re += rx * cg0; im += ry * cg0;
      }
    outv = (float)((((l1 + l2 + l3) & 1) == 0) ? re : im);
  }
  cg[combo * 125 + (a * 5 + b) * 5 + c] = outv;
}

// ===========================================================================
// Setup: path / pair tables (mirrors tp_paths / lin_pairs enumeration order)
// ===========================================================================
__device__ void build_lin(const int* iri, int ni, const int* iro, int no, int* tab) {
  int offi[40], offo[40];
  int o = 0;
  for (int s = 0; s < ni; s++) { offi[s] = o; o += iri[3*s] * (2*iri[3*s+1] + 1); }
  o = 0;
  for (int s = 0; s < no; s++) { offo[s] = o; o += iro[3*s] * (2*iro[3*s+1] + 1); }
  float fan[40];
  for (int s = 0; s < no; s++) fan[s] = 0.f;
  for (int si = 0; si < ni; si++)
    for (int so = 0; so < no; so++)
      if (iri[3*si+1] == iro[3*so+1] && iri[3*si+2] == iro[3*so+2]) fan[so] += (float)iri[3*si];
  int cnt = 0, woff = 0;
  for (int si = 0; si < ni; si++)
    for (int so = 0; so < no; so++)
      if (iri[3*si+1] == iro[3*so+1] && iri[3*si+2] == iro[3*so+2]) {
        int* e = tab + 1 + cnt * 7;
        e[0] = offi[si]; e[1] = iri[3*si]; e[2] = iri[3*si+1];
        e[3] = offo[so]; e[4] = iro[3*so]; e[5] = woff;
        e[6] = __float_as_int(1.0f / sqrtf(fan[so]));
        woff += iri[3*si] * iro[3*so];
        cnt++;
      }
  tab[0] = cnt;
}

__device__ void build_tp(const int* ir1, int n1, const int* ir2, int n2,
                         const int* iro, int no, int* tab) {
  int o1s[8], o2s[8], oos[8];
  int o = 0;
  for (int s = 0; s < n1; s++) { o1s[s] = o; o += ir1[3*s] * (2*ir1[3*s+1] + 1); }
  o = 0;
  for (int s = 0; s < n2; s++) { o2s[s] = o; o += ir2[3*s] * (2*ir2[3*s+1] + 1); }
  o = 0;
  for (int s = 0; s < no; s++) { oos[s] = o; o += iro[3*s] * (2*iro[3*s+1] + 1); }
  float fan[8];
  for (int s = 0; s < no; s++) fan[s] = 0.f;
  for (int i1 = 0; i1 < n1; i1++)
    for (int i2 = 0; i2 < n2; i2++)
      for (int io = 0; io < no; io++) {
        int l1 = ir1[3*i1+1], l2 = ir2[3*i2+1], lo = iro[3*io+1];
        int dmin = l1 - l2; if (dmin < 0) dmin = -dmin;
        if (iro[3*io+2] == ir1[3*i1+2] * ir2[3*i2+2] && lo >= dmin && lo <= l1 + l2)
          fan[io] += (float)(ir1[3*i1] * ir2[3*i2]);
      }
  int cnt = 0, woff = 0;
  for (int i1 = 0; i1 < n1; i1++)
    for (int i2 = 0; i2 < n2; i2++)
      for (int io = 0; io < no; io++) {
        int l1 = ir1[3*i1+1], l2 = ir2[3*i2+1], lo = iro[3*io+1];
        int dmin = l1 - l2; if (dmin < 0) dmin = -dmin;
        if (iro[3*io+2] == ir1[3*i1+2] * ir2[3*i2+2] && lo >= dmin && lo <= l1 + l2) {
          int* e = tab + 1 + cnt * 12;
          e[0] = o1s[i1]; e[1] = ir1[3*i1]; e[2] = l1;
          e[3] = o2s[i2]; e[4] = ir2[3*i2]; e[5] = l2;
          e[6] = oos[io]; e[7] = iro[3*io]; e[8] = lo;
          e[9] = woff;
          e[10] = __float_as_int(1.0f / sqrtf(fan[io]));
          e[11] = 0;
          woff += ir1[3*i1] * ir2[3*i2] * iro[3*io];
          cnt++;
        }
      }
  tab[0] = cnt;
}

__global__ void k_tables(int* p1, int* p2, int* pb, int* lt) {
  build_lin(IRD_IN, 1,  IRD_IN, 1,  lt + 0*512);
  build_lin(IRD_MID1, 3, IRD_MID1, 3, lt + 1*512);
  build_lin(IRD_MID2, 6, IRD_MID2, 6, lt + 2*512);
  build_lin(IRD_MID2, 6, IRD_CC, 34,  lt + 3*512);
  build_lin(IRD_MID2, 6, IRD_HH, 6,   lt + 4*512);
  build_lin(IRD_MID2, 6, IRD_CH, 13,  lt + 5*512);
  build_lin(IRD_MID2, 6, IRD_SCR, 2,  lt + 6*512);
  build_lin(IRD_SCR, 2,  IRD_SCR2, 2, lt + 7*512);
  build_lin(IRD_MID2, 6, IRD_G1, 1,   lt + 8*512);
  build_lin(IRD_G1, 1,   IRD_G2, 1,   lt + 9*512);
  build_tp(IRD_IN, 1,   IRD_SH, 3,   IRD_MID1, 3, p1);
  build_tp(IRD_MID1, 3, IRD_SH, 3,   IRD_MID2, 6, p2);
  build_tp(IRD_MID2, 6, IRD_MID2, 6, IRD_MID2, 6, pb);
}

// ===========================================================================
// Generic equivariant linear:  out = post_scale * lin(scale * tanh_head(x))
// One thread per row; zeroes its whole output row then accumulates.
// ===========================================================================
__global__ void __launch_bounds__(128) k_lin(
    const float* __restrict__ x, int indim,
    float* __restrict__ out, int outdim,
    const float* __restrict__ w,
    const int* __restrict__ tab,
    int nrows, const float* __restrict__ nnptr, int tanh_cols, float post_scale)
{
  int r = blockIdx.x * blockDim.x + threadIdx.x;
  if (r >= nrows) return;
  float scale = nnptr ? rsqrtf(nnptr[0]) : 1.0f;
  const float* xr = x + (size_t)r * indim;
  float* orow = out + (size_t)r * outdim;
  for (int c = 0; c < outdim; c++) orow[c] = 0.f;
  int np = tab[0];
  for (int p = 0; p < np; p++) {
    const int* e = tab + 1 + p * 7;
    int oi = e[0], mi = e[1], l = e[2], oo = e[3], mo = e[4], woff = e[5];
    float norm = __int_as_float(e[6]) * post_scale;
    int d = 2 * l + 1;
    for (int v = 0; v < mo; v++)
      for (int i = 0; i < d; i++) {
        float acc = 0.f;
        for (int u = 0; u < mi; u++) {
          int ix = oi + u * d + i;
          float xv = xr[ix];
          if (ix < tanh_cols) xv = tanhf(xv);
          acc += xv * scale * w[woff + u * mo + v];
        }
        orow[oo + v * d + i] += norm * acc;
      }
  }
}

// in-place tanh on the leading `cols` of each row
__global__ void k_act(float* __restrict__ x, int dim, int rows, int cols) {
  int i = blockIdx.x * blockDim.x + threadIdx.x;
  if (i >= rows * cols) return;
  int r = i / cols, c = i % cols;
  x[(size_t)r * dim + c] = tanhf(x[(size_t)r * dim + c]);
}

// ===========================================================================
// Shared device helpers: embed-MLP hidden state + WMMA weight-tile GEMM
// ===========================================================================
__device__ inline void mlp3(float eb, const float* __restrict__ wfc, float* __restrict__ hs) {
  const float* W0 = wfc;
  const float* W1 = wfc + 16;
  const float* W2 = wfc + 272;
  float h[16], g[16];
#pragma unroll
  for (int j = 0; j < 16; j++) h[j] = fmaxf(eb * W0[j], 0.f);
#pragma unroll
  for (int j = 0; j < 16; j++) {
    float s = 0.f;
#pragma unroll
    for (int k = 0; k < 16; k++) s += h[k] * W1[k * 16 + j];
    g[j] = fmaxf(0.25f * s, 0.f);
  }
#pragma unroll
  for (int j = 0; j < 16; j++) {
    float s = 0.f;
#pragma unroll
    for (int k = 0; k < 16; k++) s += g[k] * W2[k * 16 + j];
    hs[j] = fmaxf(0.25f * s, 0.f);
  }
}

// WW[0:rows, 0:cols] = 0.25 * Hs(rows x 16) @ W3[:, wcol0 : wcol0+cols]
// via v_wmma_f32_16x16x4_f32 (K=16 in 4 chained steps). N-tiles distributed
// across waves; all four K-step B fragments held in registers across the
// M-tile sweep (A fragments stream from padded LDS). WW has stride cols+1.
__device__ inline void block_gemm_ww(const float* __restrict__ Hs,
                                     const float* __restrict__ W3, int ld, int wcol0,
                                     float* __restrict__ WW, int cols,
                                     int mtiles, int tid, int nthreads)
{
  const int lane = tid & 31;
  const int wv = tid >> 5;
  const int nwv = nthreads >> 5;
  const int ntiles = cols >> 4;
  const int wwstride = cols + 1;
  const int khalf = (lane >> 4) << 1;   // K offset for this half-wave
  const int lrow = lane & 15;
  for (int nt = wv; nt < ntiles; nt += nwv) {
    const int ncol = nt * 16 + lrow;
    const float* Wc = W3 + wcol0 + ncol;
#if __has_builtin(__builtin_amdgcn_wmma_f32_16x16x4_f32)
    // hoisted B fragments for K-steps 0..3 (rows khalf, khalf+1 of each 4-row slab)
    v2f b0, b1, b2, b3;
    b0[0] = Wc[(size_t)(khalf)      * ld];  b0[1] = Wc[(size_t)(khalf + 1)  * ld];
    b1[0] = Wc[(size_t)(khalf + 4)  * ld];  b1[1] = Wc[(size_t)(khalf + 5)  * ld];
    b2[0] = Wc[(size_t)(khalf + 8)  * ld];  b2[1] = Wc[(size_t)(khalf + 9)  * ld];
    b3[0] = Wc[(size_t)(khalf + 12) * ld];  b3[1] = Wc[(size_t)(khalf + 13) * ld];
    for (int mt = 0; mt < mtiles; mt++) {
      const int mrow = mt * 16 + lrow;
      const float* Hr = Hs + mrow * HSTR + khalf;
      v8f c = {0.f, 0.f, 0.f, 0.f, 0.f, 0.f, 0.f, 0.f};
      v2f a;
      a[0] = Hr[0];  a[1] = Hr[1];
      c = __builtin_amdgcn_wmma_f32_16x16x4_f32(false, a, false, b0, (short)0, c, false, false);
      a[0] = Hr[4];  a[1] = Hr[5];
      c = __builtin_amdgcn_wmma_f32_16x16x4_f32(false, a, false, b1, (short)0, c, false, false);
      a[0] = Hr[8];  a[1] = Hr[9];
      c = __builtin_amdgcn_wmma_f32_16x16x4_f32(false, a, false, b2, (short)0, c, false, false);
      a[0] = Hr[12]; a[1] = Hr[13];
      c = __builtin_amdgcn_wmma_f32_16x16x4_f32(false, a, false, b3, (short)0, c, false, false);
#pragma unroll
      for (int i = 0; i < 8; i++) {
        const int m = mt * 16 + i + ((lane >> 4) << 3);
        WW[m * wwstride + nt * 16 + lrow] = 0.25f * c[i];
      }
    }
#else
    // VALU fallback matching the 16x16 f32 C/D fragment layout
    for (int mt = 0; mt < mtiles; mt++) {
#pragma unroll
      for (int i = 0; i < 8; i++) {
        const int m = mt * 16 + i + ((lane >> 4) << 3);
        float s = 0.f;
        for (int k = 0; k < 16; k++) s += Hs[m * HSTR + k] * Wc[(size_t)k * ld];
        WW[m * wwstride + nt * 16 + lrow] = 0.25f * s;
      }
    }
#endif
  }
}

// ===========================================================================
// Fused edge kernel: per-edge MLP -> WMMA weight tiles -> CG tensor product
// (RHS = spherical harmonics, m2 == 1) -> atomic scatter to node accumulator.
// ===========================================================================
__global__ void __launch_bounds__(EPB) k_edge_tp(
    const float* __restrict__ xnode, int xdim,
    const float* __restrict__ sh,
    const float* __restrict__ emb,
    const float* __restrict__ wfc, int ncols,
    const int* __restrict__ ptab,
    const float* __restrict__ cg,
    const int* __restrict__ esrc, const int* __restrict__ edst,
    int nedges,
    float* __restrict__ out, int outdim)
{
  extern __shared__ float smem[];
  float* Hs = smem;                 // EPB * HSTR
  float* WW = smem + EPB * HSTR;    // EPB * 65 max
  const int tid = threadIdx.x;
  const int e = blockIdx.x * EPB + tid;
  const bool valid = (e < nedges);
  const float* W3 = wfc + 528;

  float eb = valid ? emb[e] : 0.f;
  mlp3(eb, wfc, Hs + tid * HSTR);

  float bsh[9];
#pragma unroll
  for (int j = 0; j < 9; j++) bsh[j] = valid ? sh[(size_t)e * 9 + j] : 0.f;
  const int src = valid ? esrc[e] : 0;
  const int dst = valid ? edst[e] : 0;
  __syncthreads();

  const int npaths = ptab[0];
  for (int p = 0; p < npaths; p++) {
    const int* E = ptab + 1 + p * 12;
    const int o1 = E[0], m1 = E[1], l1 = E[2], o2 = E[3], l2 = E[5];
    const int oo = E[6], mo = E[7], lo = E[8], woff = E[9];
    const float norm = __int_as_float(E[10]);
    const int cols = m1 * mo;

    // speculative prefetch of next path's weight slice (global_prefetch_b8)
    if (p + 1 < npaths) {
      const int nwoff = ptab[1 + (p + 1) * 12 + 9];
      __builtin_prefetch(&W3[(size_t)(tid & 15) * ncols + nwoff + ((tid >> 4) << 3)], 0, 3);
    }

    block_gemm_ww(Hs, W3, ncols, woff, WW, cols, EPB / 16, tid, EPB);
    __syncthreads();

    if (valid) {
      const int d1 = 2*l1 + 1, d2 = 2*l2 + 1, d3 = 2*lo + 1;
      const float* C = cg + ((l1 * 3 + l2) * 3 + lo) * 125;
      float tmp[5][5];
#pragma unroll
      for (int i = 0; i < 5; i++) { if (i < d1) {
#pragma unroll
        for (int k = 0; k < 5; k++) { if (k < d3) {
          float t = 0.f;
#pragma unroll
          for (int j = 0; j < 5; j++) { if (j < d2) t += bsh[o2 + j] * C[(i*5 + j)*5 + k]; }
          tmp[i][k] = t;
        } }
      } }
      float s[8][5];
      const float* arow = xnode + (size_t)src * xdim + o1;
#pragma unroll
      for (int u = 0; u < 8; u++) { if (u < m1) {
#pragma unroll
        for (int k = 0; k < 5; k++) { if (k < d3) {
          float t = 0.f;
#pragma unroll
          for (int i = 0; i < 5; i++) { if (i < d1) t += arow[u*d1 + i] * tmp[i][k]; }
          s[u][k] = t;
        } }
      } }
      float* orow = out + (size_t)dst * outdim + oo;
      const float* wwr = WW + tid * (cols + 1);
      for (int w = 0; w < mo; w++) {
#pragma unroll
        for (int k = 0; k < 5; k++) { if (k < d3) {
          float t = 0.f;
#pragma unroll
          for (int u = 0; u < 8; u++) { if (u < m1) t += wwr[u*mo + w] * s[u][k]; }
          unsafeAtomicAdd(&orow[w*d3 + k], norm * t);
        } }
      }
    }
    __syncthreads();
  }
}

// ===========================================================================
// Fused bond kernel: full MID2 (x) MID2 -> MID2 tensor product, 60 paths,
// weights streamed in 128-col chunks through LDS WMMA tiles.
// ===========================================================================
__global__ void __launch_bounds__(BPB) k_bond_tp(
    const float* __restrict__ x3,
    const float* __restrict__ emb,
    const float* __restrict__ wfc, int ncols,
    const int* __restrict__ ptab,
    const float* __restrict__ cg,
    const int* __restrict__ esrc, const int* __restrict__ edst,
    const int* __restrict__ bind, int nbond,
    float* __restrict__ outF)
{
  extern __shared__ float smem[];
  float* Hs = smem;                 // BPB * HSTR
  float* WW = smem + BPB * HSTR;    // BPB * 129
  const int tid = threadIdx.x;
  const int t = blockIdx.x * BPB + tid;
  const bool valid = (t < nbond);
  const int be = valid ? bind[t] : 0;
  const int src = valid ? esrc[be] : 0;
  const int dst = valid ? edst[be] : 0;
  const float* W3 = wfc + 528;

  float eb = valid ? emb[be] : 0.f;
  mlp3(eb, wfc, Hs + tid * HSTR);
  __syncthreads();

  const float* arow0 = x3 + (size_t)src * 144;
  const float* brow0 = x3 + (size_t)dst * 144;
  const int npaths = ptab[0];

  for (int p = 0; p < npaths; p++) {
    const int* E = ptab + 1 + p * 12;
    const int o1 = E[0], l1 = E[2], o2 = E[3], l2 = E[5];
    const int oo = E[6], lo = E[8], woff = E[9];
    const float norm = __int_as_float(E[10]);
    const int d1 = 2*l1 + 1, d2 = 2*l2 + 1, d3 = 2*lo + 1;
    const float* C = cg + ((l1 * 3 + l2) * 3 + lo) * 125;

    float acc[8][5];
#pragma unroll
    for (int w = 0; w < 8; w++)
#pragma unroll
      for (int k = 0; k < 5; k++) acc[w][k] = 0.f;

    for (int ch = 0; ch < 4; ch++) {            // u in {2ch, 2ch+1}, 128 cols
      // speculative prefetch of the next 128-col weight chunk
      {
        int next_off = (ch < 3) ? (woff + (ch + 1) * 128)
                     : ((p + 1 < npaths) ? ptab[1 + (p + 1) * 12 + 9] : -1);
        if (next_off >= 0)
          __builtin_prefetch(&W3[(size_t)(tid & 15) * ncols + next_off + (((tid >> 4) & 3) << 5)], 0, 3);
      }
      block_gemm_ww(Hs, W3, ncols, woff + ch * 128, WW, 128, BPB / 16, tid, BPB);
      __syncthreads();
      if (valid) {
        const float* wwr = WW + tid * 129;
        for (int v = 0; v < 8; v++) {
          float tv[5][5];
#pragma unroll
          for (int i = 0; i < 5; i++) { if (i < d1) {
#pragma unroll
            for (int k = 0; k < 5; k++) { if (k < d3) {
              float s = 0.f;
#pragma unroll
              for (int j = 0; j < 5; j++) { if (j < d2) s += brow0[o2 + v*d2 + j] * C[(i*5 + j)*5 + k]; }
              tv[i][k] = s;
            } }
          } }
          for (int uu = 0; uu < 2; uu++) {
            const int u = ch * 2 + uu;
            float sv[5];
#pragma unroll
            for (int k = 0; k < 5; k++) { if (k < d3) {
              float s = 0.f;
#pragma unroll
              for (int i = 0; i < 5; i++) { if (i < d1) s += arow0[o1 + u*d1 + i] * tv[i][k]; }
              sv[k] = s;
            } }
#pragma unroll
            for (int w = 0; w < 8; w++) {
              const float wwv = wwr[(uu * 8 + v) * 8 + w];
#pragma unroll
              for (int k = 0; k < 5; k++) { if (k < d3) acc[w][k] += wwv * sv[k]; }
            }
          }
        }
      }
      __syncthreads();
    }
    if (valid) {
      float* orow = outF + (size_t)t * 144 + oo;
#pragma unroll
      for (int w = 0; w < 8; w++)
#pragma unroll
        for (int k = 0; k < 5; k++) { if (k < d3) orow[w*d3 + k] += norm * acc[w][k]; }
    }
  }
}

// ===========================================================================
// Host orchestration (graph-capture safe: kernels + memset nodes only)
// ===========================================================================
extern "C" void kernel_launch(void* const* d_in, const int* in_sizes, int n_in,
                              void* d_out, int out_size, void* d_ws, size_t ws_size,
                              hipStream_t stream) {
  (void)in_sizes; (void)n_in; (void)out_size; (void)ws_size;
  const float* f_in  = (const float*)d_in[0];
  const float* sh    = (const float*)d_in[1];
  const float* emb   = (const float*)d_in[2];
  const float* nn    = (const float*)d_in[3];
  const float* w_lin1 = (const float*)d_in[4];
  const float* w_lin2 = (const float*)d_in[5];
  const float* w_lin3 = (const float*)d_in[6];
  const float* w_linCC = (const float*)d_in[7];
  const float* w_linHH = (const float*)d_in[8];
  const float* w_linCH = (const float*)d_in[9];
  const float* w_linC  = (const float*)d_in[10];
  const float* w_linH  = (const float*)d_in[11];
  const float* w_scr1  = (const float*)d_in[12];
  const float* w_scr2  = (const float*)d_in[13];
  const float* w_gap1  = (const float*)d_in[14];
  const float* w_gap2  = (const float*)d_in[15];
  const float* w_fc1   = (const float*)d_in[16];
  const float* w_fc2   = (const float*)d_in[17];
  const float* w_fcb   = (const float*)d_in[18];
  const int* esrc  = (const int*)d_in[19];
  const int* edst  = (const int*)d_in[20];
  const int* HHind = (const int*)d_in[21];
  const int* CCind = (const int*)d_in[22];
  const int* CHind = (const int*)d_in[23];

  float* ws  = (float*)d_ws;
  float* cgp = ws + WS_CG;
  int* p1 = (int*)(ws + WS_P1);
  int* p2 = (int*)(ws + WS_P2);
  int* pb = (int*)(ws + WS_PB);
  int* lt = (int*)(ws + WS_LT);
  float* x1  = ws + WS_X1;
  float* xn1 = ws + WS_XN1;
  float* x2  = ws + WS_X2;
  float* xn2 = ws + WS_XN2;
  float* x3  = ws + WS_X3;
  float* HHf = ws + WS_HHF;
  float* CCf = ws + WS_CCF;
  float* CHf = ws + WS_CHF;
  float* scr1 = ws + WS_SCR;
  float* g1b  = ws + WS_G1B;
  float* out = (float*)d_out;

  const size_t EDGE_SMEM = (size_t)(EPB * HSTR + EPB * 65) * 4;
  const size_t BOND_SMEM = (size_t)(BPB * HSTR + BPB * 129) * 4;

  // ---- setup: CG tensors + path/pair tables (rebuilt every call) ----
  k_cg<<<dim3((27 * 125 + 127) / 128), 128, 0, stream>>>(cgp);
  k_tables<<<1, 1, 0, stream>>>(p1, p2, pb, lt);
  hipMemsetAsync(xn1, 0, (size_t)N_NODES * 72 * 4, stream);
  hipMemsetAsync(xn2, 0, (size_t)N_NODES * 144 * 4, stream);
  hipMemsetAsync(HHf, 0, (size_t)3 * N_BOND * 144 * 4, stream);   // HHf,CCf,CHf contiguous

  auto linL = [&](const float* x, int indim, float* o, int outdim, const float* w,
                  int tabidx, int rows, const float* nnp, int tanhc, float ps) {
    k_lin<<<dim3((rows + 127) / 128), 128, 0, stream>>>(
        x, indim, o, outdim, w, lt + tabidx * 512, rows, nnp, tanhc, ps);
  };

  // ---- layer 0: x = tanh(lin(f_in)) ----
  linL(f_in, 2, x1, 2, w_lin1, 0, N_NODES, nullptr, 0, 1.f);
  k_act<<<dim3((N_NODES * 2 + 127) / 128), 128, 0, stream>>>(x1, 2, N_NODES, 2);

  // ---- message pass 1: TP(IN x SH -> MID1), scatter-sum ----
  k_edge_tp<<<dim3((N_EDGES + EPB - 1) / EPB), EPB, EDGE_SMEM, stream>>>(
      x1, 2, sh, emb, w_fc1, 48, p1, cgp, esrc, edst, N_EDGES, xn1, 72);
  linL(xn1, 72, x2, 72, w_lin2, 1, N_NODES, nn, 0, 1.f);
  k_act<<<dim3((N_NODES * 8 + 127) / 128), 128, 0, stream>>>(x2, 72, N_NODES, 8);

  // ---- message pass 2: TP(MID1 x SH -> MID2), scatter-sum ----
  k_edge_tp<<<dim3((N_EDGES + EPB - 1) / EPB), EPB, EDGE_SMEM, stream>>>(
      x2, 72, sh, emb, w_fc2, 960, p2, cgp, esrc, edst, N_EDGES, xn2, 144);
  linL(xn2, 144, x3, 144, w_lin3, 2, N_NODES, nn, 0, 1.f);
  k_act<<<dim3((N_NODES * 16 + 127) / 128), 128, 0, stream>>>(x3, 144, N_NODES, 16);

  // ---- bond features: TP(MID2 x MID2 -> MID2) for HH / CC / CH ----
  const dim3 bg((N_BOND + BPB - 1) / BPB);
  k_bond_tp<<<bg, BPB, BOND_SMEM, stream>>>(x3, emb, w_fcb, 30720, pb, cgp,
                                            esrc, edst, HHind, N_BOND, HHf);
  k_bond_tp<<<bg, BPB, BOND_SMEM, stream>>>(x3, emb, w_fcb, 30720, pb, cgp,
                                            esrc, edst, CCind, N_BOND, CCf);
  k_bond_tp<<<bg, BPB, BOND_SMEM, stream>>>(x3, emb, w_fcb, 30720, pb, cgp,
                                            esrc, edst, CHind, N_BOND, CHf);

  // ---- heads (tanh on first 16 cols fused where ACT3 applies) ----
  linL(HHf, 144, out + OUT_eHH, 25,  w_linHH, 4, N_BOND, nullptr, 16, SCALING);
  linL(CHf, 144, out + OUT_eCH, 70,  w_linCH, 5, N_BOND, nullptr, 16, SCALING);
  linL(CCf, 144, out + OUT_eCC, 196, w_linCC, 3, N_BOND, nullptr, 16, SCALING);
  linL(x3, 144, out + OUT_nC, 196, w_linC, 3, N_NODES, nullptr, 0, SCALING);
  linL(x3, 144, out + OUT_nH, 25,  w_linH, 4, N_NODES, nullptr, 0, SCALING);

  // ---- screen: lin -> tanh(0e head) -> lin ----
  linL(x3, 144, scr1, 37, w_scr1, 6, N_NODES, nullptr, 0, 1.f);
  linL(scr1, 37, out + OUT_screen, 6, w_scr2, 7, N_NODES, nullptr, 32, 1.f);

  // ---- gaps: lin -> tanh(all 32) -> lin, on raw bond features ----
  linL(CCf, 144, g1b, 32, w_gap1, 8, N_BOND, nullptr, 0, 1.f);
  linL(g1b, 32, out + OUT_gCC, 3, w_gap2, 9, N_BOND, nullptr, 32, 1.f);
  linL(HHf, 144, g1b, 32, w_gap1, 8, N_BOND, nullptr, 0, 1.f);
  linL(g1b, 32, out + OUT_gHH, 3, w_gap2, 9, N_BOND, nullptr, 32, 1.f);
  linL(CHf, 144, g1b, 32, w_gap1, 8, N_BOND, nullptr, 0, 1.f);
  linL(g1b, 32, out + OUT_gCH, 3, w_gap2, 9, N_BOND, nullptr, 32, 1.f);
}